// NeuronClassPyTorch_3126736191737
// MI455X (gfx1250) — compile-verified
//
#include <hip/hip_runtime.h>

// ---------------------------------------------------------------------------
// Online-learning LIF neuron recurrence, MI455X (gfx1250, wave32, WMMA).
//
// Per step t:   xw_b = sum_n x[t,b,n] w[n]
//               v = a*v + xw - z ;  z = (v>1) ;  E = xw - v*(w.w)
//               w += (eta/B) * ( x^T v + p^T E - (sum v^2) * w )
//               p  = a*p + x_t
//
// 16 persistent WGs, each owns 128 of N=2048.  p-slice lives in LDS; the x_t
// slice is DOUBLE-BUFFERED in LDS and fetched with GLOBAL_LOAD_ASYNC_TO_LDS
// (ASYNCcnt) so step t+1's HBM traffic overlaps step t's WMMA + grid barrier.
// One 129-float global partial + one atomic-counter grid barrier per step.
// All dense reductions via V_WMMA_F32_16X16X4_F32 (f32 accumulate).
// ---------------------------------------------------------------------------

#define Tn   500
#define Bn   128
#define Nn   2048
#define NWG  16
#define NSL  128           // N-slice per workgroup
#define XST  132           // LDS row stride of x slice (bank-conflict pad)
#define PROW 132           // floats per WG in partial buffer (128 xw + 1 ww)
#define XBUFF (Bn * XST)   // floats per x buffer

typedef float v2f __attribute__((ext_vector_type(2)));
typedef float v8f __attribute__((ext_vector_type(8)));

__global__ void __launch_bounds__(256, 1)
neuron_init(int* cnt) {
    int i = blockIdx.x * blockDim.x + threadIdx.x;
    if (i < Tn) cnt[i] = 0;
}

// Issue this wave's share (16 instrs) of one 64KB x-slice: global -> LDS,
// no VGPR round trip.  GVS mode: vdst = per-lane LDS byte addr, vaddr =
// per-lane 32-bit byte offset, saddr = uniform 64-bit base.
__device__ __forceinline__ void stage_x_async(const float* xt,
                                              unsigned int lds_base, int tid) {
    #pragma unroll
    for (int k = 0; k < 16; ++k) {
        int i = tid + (k << 8);          // 4096 x b128 chunks total
        int b = i >> 5, q = i & 31;      // row (batch), 16B chunk in row
        unsigned int voff = (unsigned int)(b * (Nn * 4) + q * 16);
        unsigned int lds  = lds_base + (unsigned int)(b * (XST * 4) + q * 16);
        asm volatile("global_load_async_to_lds_b128 %0, %1, %2"
                     :: "v"(lds), "v"(voff), "s"(xt) : "memory");
    }
}

__global__ void __launch_bounds__(256, 1)
neuron_persistent(const float* __restrict__ x, const float* __restrict__ w0,
                  float* __restrict__ out, int* __restrict__ cnt,
                  float* __restrict__ part) {
    extern __shared__ float sm[];
    float* xbufA = sm;                   // Bn * XST  (x slice, even t)
    float* xbufB = xbufA + XBUFF;        // Bn * XST  (x slice, odd t)
    float* ps    = xbufB + XBUFF;        // Bn * NSL  (p slice)
    float* ws    = ps    + Bn * NSL;     // NSL       (w slice)
    float* vsh   = ws    + NSL;          // Bn  (replicated v)
    float* zsh   = vsh   + Bn;           // Bn  (replicated z)
    float* esh   = zsh   + Bn;           // Bn  (replicated E)
    float* red   = esh   + Bn;           // Bn  (reduction scratch)

    const int tid  = threadIdx.x;
    const int wg   = blockIdx.x;
    const int lane = tid & 31;
    const int wave = tid >> 5;
    const int l15  = lane & 15;
    const int hi   = lane >> 4;          // 0: lanes 0-15, 1: lanes 16-31
    const int n0   = wg * NSL;

    const unsigned int baseA = (unsigned int)(unsigned long long)xbufA;
    const unsigned int baseB = (unsigned int)(unsigned long long)xbufB;

    const float ALPHA = 0.9f;            // 1 - DT/TAU_M
    const float VTH   = 1.0f;
    const float ETAB  = 1e-4f / 128.0f;  // ETA / BATCH

    // ---- initial state: w slice from w0; v=z=0; p=0 -----------------------
    if (tid < NSL) ws[tid] = w0[n0 + tid];
    if (tid < Bn)  { vsh[tid] = 0.f; zsh[tid] = 0.f; }
    for (int i = tid; i < Bn * NSL; i += 256) ps[i] = 0.f;

    // Prologue: kick off async fetch of x_0 slice into buffer A.
    stage_x_async(x + n0, baseA, tid);

    __syncthreads();

    for (int t = 0; t < Tn; ++t) {
        const float* xb = (t & 1) ? xbufB : xbufA;   // this step's slice
        const unsigned int nbase = (t & 1) ? baseA : baseB;

        // ---- issue async fetch of x_{t+1} into the other buffer, then ----
        // ---- wait for THIS step's 16 loads (async loads retire in order) -
        if (t + 1 < Tn) {
            stage_x_async(x + (size_t)(t + 1) * (Bn * Nn) + n0, nbase, tid);
            asm volatile("s_wait_asynccnt 0x10" ::: "memory");
        } else {
            asm volatile("s_wait_asynccnt 0x0" ::: "memory");
        }
        if (tid < NSL) red[tid] = ws[tid] * ws[tid];   // for local w.w
        __syncthreads();                 // all waves' async writes visible

        // ---- phase 2: xw partials via WMMA f32 16x16x4 --------------------
        // A(16x4) = x rows (wave's 16 batches), B(4x16) = w[k] replicated
        // across N.  A/B VGPR0 hold K={0,2} across lane halves, VGPR1 K={1,3}.
        v8f acc = {};
        {
            const float* arow = &xb[(wave * 16 + l15) * XST];
            const int ko = 2 * hi;
            for (int c = 0; c < NSL; c += 4) {
                v2f a  = *(const v2f*)(arow + c + ko);
                v2f bb = *(const v2f*)(&ws[c + ko]);
                acc = __builtin_amdgcn_wmma_f32_16x16x4_f32(
                    false, a, false, bb, (short)0, acc, false, false);
            }
        }
        float* pb = part + (size_t)(t & 1) * (NWG * PROW) + wg * PROW;
        if (l15 == 0) {                       // lanes 0 (M=k) and 16 (M=k+8)
            int base = wave * 16 + hi * 8;
            #pragma unroll
            for (int k = 0; k < 8; ++k) pb[base + k] = acc[k];
        }
        if (tid == 0) {                       // w.w partial (serial, 128 adds)
            float ss = 0.f;
            for (int j = 0; j < NSL; ++j) ss += red[j];
            pb[128] = ss;
        }

        // ---- grid barrier t (release partials, acquire all) ---------------
        __threadfence();
        __syncthreads();
        if (tid == 0) {
            __hip_atomic_fetch_add(&cnt[t], 1, __ATOMIC_RELEASE,
                                   __HIP_MEMORY_SCOPE_AGENT);
            while (__hip_atomic_load(&cnt[t], __ATOMIC_ACQUIRE,
                                     __HIP_MEMORY_SCOPE_AGENT) < NWG)
                __builtin_amdgcn_s_sleep(2);
        }
        __syncthreads();
        __threadfence();

        // ---- phase 4: v, z, E (replicated in every WG) --------------------
        if (tid < Bn) {
            const float* pp = part + (size_t)(t & 1) * (NWG * PROW);
            float xw = 0.f, ww = 0.f;
            #pragma unroll
            for (int g = 0; g < NWG; ++g) {
                xw += pp[g * PROW + tid];
                ww += pp[g * PROW + 128];
            }
            float vn = ALPHA * vsh[tid] + xw - VTH * zsh[tid];
            zsh[tid] = (vn - VTH > 0.f) ? 1.f : 0.f;
            vsh[tid] = vn;
            esh[tid] = xw - vn * ww;
            red[tid] = vn * vn;
            if (wg == 0) out[(size_t)t * Bn + tid] = vn;   // voltage trace
        }
        __syncthreads();
        for (int off = 64; off > 0; off >>= 1) {           // s = sum v^2
            if (tid < off) red[tid] += red[tid + off];
            __syncthreads();
        }
        float s = red[0];

        // ---- phase 5a: vec1 = x^T v, vec2 = p^T E via WMMA ----------------
        // A(16x4) = v/E[bc+k] broadcast across M; B(4x16) = x/p rows
        // (K = batch, N = this wave's 16 n-columns).
        v8f a1 = {}, a2 = {};
        {
            const int jn = wave * 16 + l15;
            const int ko = 2 * hi;
            for (int bc = 0; bc < Bn; bc += 4) {
                v2f av = *(const v2f*)(&vsh[bc + ko]);
                v2f ae = *(const v2f*)(&esh[bc + ko]);
                v2f bx, bp;
                bx.x = xb[(bc + ko)     * XST + jn];
                bx.y = xb[(bc + ko + 1) * XST + jn];
                bp.x = ps[(bc + ko)     * NSL + jn];
                bp.y = ps[(bc + ko + 1) * NSL + jn];
                a1 = __builtin_amdgcn_wmma_f32_16x16x4_f32(
                    false, av, false, bx, (short)0, a1, false, false);
                a2 = __builtin_amdgcn_wmma_f32_16x16x4_f32(
                    false, ae, false, bp, (short)0, a2, false, false);
            }
        }
        __syncthreads();   // all vec2 reads of p done before p is rewritten

        // ---- phase 5b: w update (lanes 0-15 own one n each), p update -----
        if (lane < 16) {
            int nl = wave * 16 + lane;
            ws[nl] += ETAB * (a1[0] + a2[0] - s * ws[nl]);
        }
        for (int i = tid; i < Bn * NSL; i += 256) {
            int b = i >> 7, nl = i & 127;
            ps[i] = ALPHA * ps[i] + xb[b * XST + nl];
        }
        __syncthreads();   // ws/ps stable, xb reads done, before next issue
    }
}

extern "C" void kernel_launch(void* const* d_in, const int* in_sizes, int n_in,
                              void* d_out, int out_size, void* d_ws, size_t ws_size,
                              hipStream_t stream) {
    (void)in_sizes; (void)n_in; (void)out_size; (void)ws_size;
    const float* x  = (const float*)d_in[0];   // (T, B, N) f32
    const float* w0 = (const float*)d_in[1];   // (N,) f32
    float* out = (float*)d_out;                // (T, B) f32

    int*   cnt  = (int*)d_ws;                          // Tn counters
    float* part = (float*)((char*)d_ws + 4096);        // 2 x NWG x PROW f32

    const size_t smem =
        (size_t)(2 * XBUFF + Bn * NSL + NSL + 4 * Bn) * sizeof(float);
    // Allow >64KB dynamic LDS (gfx1250: 320KB/WGP). Ignore error if unneeded.
    static_cast<void>(hipFuncSetAttribute(
        reinterpret_cast<const void*>(neuron_persistent),
        hipFuncAttributeMaxDynamicSharedMemorySize, (int)smem));

    neuron_init<<<(Tn + 255) / 256, 256, 0, stream>>>(cnt);
    neuron_persistent<<<NWG, 256, smem, stream>>>(x, w0, out, cnt, part);
}